// OrdinalContrastiveLoss_mm_82970178224609
// MI455X (gfx1250) — compile-verified
//
#include <hip/hip_runtime.h>
#include <hip/hip_bf16.h>

#define NROW 512
#define DDIM 512
#define NCLS 10
#define EPSF 1e-8f

typedef __attribute__((ext_vector_type(2))) float v2f;
typedef __attribute__((ext_vector_type(8))) float v8f;

// ---------------------------------------------------------------------------
// Kernel 1: row L2-normalization. One 256-thread block per row.
// ---------------------------------------------------------------------------
__global__ void ocl_normalize_rows(const float* __restrict__ pred,
                                   float* __restrict__ xn) {
  const int row = blockIdx.x;
  const int t   = threadIdx.x;
  const float* x = pred + row * DDIM;
  float a0 = x[t];
  float a1 = x[t + 256];
  __shared__ float red[256];
  red[t] = a0 * a0 + a1 * a1;
  __syncthreads();
  for (int s = 128; s > 0; s >>= 1) {
    if (t < s) red[t] += red[t + s];
    __syncthreads();
  }
  const float norm  = sqrtf(red[0]);
  const float scale = 1.0f / fmaxf(norm, EPSF);
  xn[row * DDIM + t]       = a0 * scale;
  xn[row * DDIM + t + 256] = a1 * scale;
}

// ---------------------------------------------------------------------------
// Kernel 2: softplus(distances_raw), cumsum -> class_pos[10]. Tiny; 1 thread.
// ---------------------------------------------------------------------------
__global__ void ocl_class_pos(const float* __restrict__ draw,
                              float* __restrict__ cpos) {
  if (threadIdx.x == 0 && blockIdx.x == 0) {
    float acc = 0.0f;
    cpos[0] = 0.0f;
    for (int i = 0; i < NCLS - 1; ++i) {
      float x  = draw[i];
      float sp = (x > 0.0f) ? (x + log1pf(expf(-x))) : log1pf(expf(x));
      acc += sp;
      cpos[i + 1] = acc;
    }
  }
}

// ---------------------------------------------------------------------------
// Kernel 3: cos = xn * xn^T via V_WMMA_F32_16X16X4_F32.
// One wave (32 threads) per 16x16 output tile; K-loop in steps of 4.
//   A 16x4 f32 layout: lanes 0-15 -> M=lane, VGPR0/1 = K=0/1;
//                      lanes 16-31 -> K=2/3.  (contiguous float pair per lane)
//   B 4x16 mirrors A with N=lane since B(k,n) = xn[n][k].
//   C/D: VGPR r, lane L: element (M = r + 8*(L>=16), N = L&15).
// ---------------------------------------------------------------------------
__global__ void ocl_wmma_gemm_xxt(const float* __restrict__ xn,
                                  float* __restrict__ cosm) {
  const int tileN = blockIdx.x;
  const int tileM = blockIdx.y;
  const int lane  = threadIdx.x;      // 0..31 (wave32)
  const int half  = lane >> 4;        // 0 or 1
  const int l     = lane & 15;

  const float* rowA = xn + (tileM * 16 + l) * DDIM;
  const float* rowB = xn + (tileN * 16 + l) * DDIM;

  v8f c = {};
  for (int k = 0; k < DDIM; k += 4) {
    const int ko = k + 2 * half;
    v2f a, b;
    a.x = rowA[ko];
    a.y = rowA[ko + 1];
    b.x = rowB[ko];
    b.y = rowB[ko + 1];
    // (neg_a, A, neg_b, B, c_mod, C, reuse_a, reuse_b)
    c = __builtin_amdgcn_wmma_f32_16x16x4_f32(false, a, false, b,
                                              (short)0, c, false, false);
  }

  const int n     = tileN * 16 + l;
  const int mbase = tileM * 16 + 8 * half;
#pragma unroll
  for (int r = 0; r < 8; ++r) {
    cosm[(mbase + r) * NROW + n] = c[r];
  }
}

// ---------------------------------------------------------------------------
// Kernel 4: per-row loss. One 256-thread block per row i.
//   S[i,k] = sum_j neg[i,j] * relu(cos[i,j] + margin[i,j] - cos[i,k])
//   tot[i] = (sum_{k pos} S[i,k] / max(n_neg,1)) / max(n_pos*n_neg, 1)
// ---------------------------------------------------------------------------
__global__ void ocl_row_loss(const float* __restrict__ cosm,
                             const int* __restrict__ target,
                             const float* __restrict__ cpos,
                             float* __restrict__ tot) {
  const int i = blockIdx.x;
  const int t = threadIdx.x;

  __shared__ float s_cos[NROW];
  __shared__ float s_cj[NROW];    // cos + margin (only meaningful where neg)
  __shared__ float s_negf[NROW];  // 1.0 if negative pair, else 0.0
  __shared__ float s_red[256];
  __shared__ int   s_ired[256];

  const int   ti  = target[i];
  const float cpi = cpos[ti];

  int negcnt = 0;
  for (int j = t; j < NROW; j += 256) {
    const int   tj   = target[j];
    const float cv   = cosm[i * NROW + j];
    const int   neg  = (tj != ti) ? 1 : 0;
    const float marg = neg ? fabsf(cpi - cpos[tj]) : 0.0f;
    s_cos[j]  = cv;
    s_cj[j]   = cv + marg;
    s_negf[j] = (float)neg;
    negcnt += neg;
  }
  s_ired[t] = negcnt;
  __syncthreads();
  for (int s = 128; s > 0; s >>= 1) {
    if (t < s) s_ired[t] += s_ired[t + s];
    __syncthreads();
  }
  const int n_neg = s_ired[0];
  const int n_pos = (NROW - n_neg) - 1;
  const float inv_nneg = 1.0f / fmaxf((float)n_neg, 1.0f);

  float acc = 0.0f;
  for (int k = t; k < NROW; k += 256) {
    const bool is_pos = (k != i) && (s_negf[k] == 0.0f);
    if (is_pos) {
      const float ck = s_cos[k];
      float S = 0.0f;
      for (int j = 0; j < NROW; ++j) {
        const float d = s_cj[j] - ck;
        S += fmaxf(d, 0.0f) * s_negf[j];
      }
      // loss_ik * (count>0): if n_neg==0 then S==0 anyway.
      acc += S * inv_nneg;
    }
  }
  s_red[t] = acc;
  __syncthreads();
  for (int s = 128; s > 0; s >>= 1) {
    if (t < s) s_red[t] += s_red[t + s];
    __syncthreads();
  }
  if (t == 0) {
    const float denom = fmaxf((float)(n_pos * n_neg), 1.0f);
    tot[i] = s_red[0] / denom;
  }
}

// ---------------------------------------------------------------------------
// Kernel 5: deterministic final mean over the 512 per-row values.
// ---------------------------------------------------------------------------
__global__ void ocl_final_reduce(const float* __restrict__ tot,
                                 float* __restrict__ out) {
  const int t = threadIdx.x;
  __shared__ float red[256];
  red[t] = tot[t] + tot[t + 256];
  __syncthreads();
  for (int s = 128; s > 0; s >>= 1) {
    if (t < s) red[t] += red[t + s];
    __syncthreads();
  }
  if (t == 0) out[0] = red[0] / (float)NROW;
}

// ---------------------------------------------------------------------------
extern "C" void kernel_launch(void* const* d_in, const int* in_sizes, int n_in,
                              void* d_out, int out_size, void* d_ws, size_t ws_size,
                              hipStream_t stream) {
  const float* prediction = (const float*)d_in[0];   // 512*512 f32
  const float* dist_raw   = (const float*)d_in[1];   // 9 f32
  const int*   target     = (const int*)d_in[2];     // 512 i32
  float*       out        = (float*)d_out;           // 1 f32

  // Workspace layout (floats): xn | cos | class_pos(16) | tot(512)
  float* xn   = (float*)d_ws;
  float* cosm = xn + NROW * DDIM;
  float* cpos = cosm + NROW * NROW;
  float* tot  = cpos + 16;

  ocl_normalize_rows<<<NROW, 256, 0, stream>>>(prediction, xn);
  ocl_class_pos<<<1, 32, 0, stream>>>(dist_raw, cpos);
  {
    dim3 grid(NROW / 16, NROW / 16);
    ocl_wmma_gemm_xxt<<<grid, 32, 0, stream>>>(xn, cosm);
  }
  ocl_row_loss<<<NROW, 256, 0, stream>>>(cosm, target, cpos, tot);
  ocl_final_reduce<<<1, 256, 0, stream>>>(tot, out);
}